// L2XGCN_69157563400533
// MI455X (gfx1250) — compile-verified
//
#include <hip/hip_runtime.h>
#include <hip/hip_bf16.h>
#include <math.h>

// Problem constants (fixed by the reference)
#define GG   1024            // graphs
#define NPG  64              // nodes per graph
#define MM   512             // undirected edges per graph
#define NN   (GG * NPG)      // 65536 nodes
#define EE   (GG * 2 * MM)   // 1048576 directed edges
#define FF   128             // feature width (F_IN == HID)
#define CLS  10

typedef float v2f __attribute__((ext_vector_type(2)));
typedef float v8f __attribute__((ext_vector_type(8)));

// ---------------------------------------------------------------------------
// GEMM via V_WMMA_F32_16X16X4_F32: C[M x 128] = A[M x 128] @ W[128 x 128].
// W (64 KB) is staged in LDS once per 8-wave block. Each wave owns a 16x128
// output strip: 8 f32 accumulators, one A-fragment per k-step feeds 8 WMMAs
// (8x A reuse), B-fragments come from LDS. 256 WMMAs per wave.
// Fragment layouts per ISA: A 16x4 -> VGPR0=K{k,k+2}, VGPR1=K{k+1,k+3} split
// by lane half; C/D: M = vgpr + 8*(lane>=16), N = lane&15.
// ---------------------------------------------------------------------------
__global__ void gemm128_wmma_kernel(const float* __restrict__ A,
                                    const float* __restrict__ W,
                                    const float* __restrict__ bias,
                                    float* __restrict__ C,
                                    int Mrows, int fuse_bias_relu) {
    __shared__ float lw[FF * FF];            // 64 KB staged weights
    const int tid = threadIdx.x;
    {
        const float4* Wv = reinterpret_cast<const float4*>(W);
        float4* lv = reinterpret_cast<float4*>(lw);
#pragma unroll
        for (int i = 0; i < (FF * FF / 4) / 256; ++i)   // 16 iterations
            lv[tid + i * 256] = Wv[tid + i * 256];
    }
    __syncthreads();

    const int wv   = tid >> 5;
    const int lane = tid & 31;
    const int strip = blockIdx.x * 8 + wv;   // 16-row output strip
    if (strip * 16 >= Mrows) return;

    const int l15 = lane & 15;
    const int hi  = lane >> 4;               // 0 => K {k,k+1}; 1 => {k+2,k+3}
    const float* __restrict__ ap = A + (size_t)(strip * 16 + l15) * FF;

    v8f acc[8] = {};
    for (int k = 0; k < FF; k += 4) {
        v2f a;
        a.x = ap[k + 2 * hi];
        a.y = ap[k + 2 * hi + 1];
        const float* wp = lw + (k + 2 * hi) * FF + l15;
#pragma unroll
        for (int tn = 0; tn < 8; ++tn) {
            v2f b;
            b.x = wp[tn * 16];
            b.y = wp[tn * 16 + FF];
            acc[tn] = __builtin_amdgcn_wmma_f32_16x16x4_f32(
                false, a, false, b, (short)0, acc[tn], false, false);
        }
    }

#pragma unroll
    for (int tn = 0; tn < 8; ++tn) {
        const int col = tn * 16 + l15;
        const float bval = fuse_bias_relu ? bias[col] : 0.0f;
        float* cp = C + (size_t)(strip * 16 + hi * 8) * FF + col;
#pragma unroll
        for (int r = 0; r < 8; ++r) {
            float v = acc[tn][r] + bval;
            if (fuse_bias_relu) v = fmaxf(v, 0.0f);
            cp[(size_t)r * FF] = v;
        }
    }
}

// ---------------------------------------------------------------------------
// Fused GCN propagate: one workgroup per graph. All of a graph's state
// (64 nodes x 128 feats = 32 KB accumulator + 64 degrees) lives in LDS.
//   pass 1: deg[d] += w          (thread per edge, ds_add_f32)
//   pass 2: dinv = rsqrt(deg+1)
//   pass 3: acc[d] += dinv[s]*w*dinv[d] * h[s]  (wave per edge; lane owns
//           feats {lane, lane+32, lane+64, lane+96} -> conflict-free banks)
//   pass 4: out = [relu](acc + dinv^2 * h + bias)
// Zero global atomics, zero global scratch for aggregation.
// ---------------------------------------------------------------------------
__global__ void gcn_layer_kernel(const float* __restrict__ h,
                                 const int* __restrict__ srcv,
                                 const int* __restrict__ dstv,
                                 const float* __restrict__ ew,   // null => 1
                                 const float* __restrict__ bias,
                                 float* __restrict__ out, int relu) {
    __shared__ float acc[NPG * FF];   // 32 KB
    __shared__ float degl[NPG];
    __shared__ float dinvl[NPG];
    const int g = blockIdx.x, tid = threadIdx.x;
    const int ebase = g * (2 * MM);
    const int nbase = g * NPG;

    for (int t = tid; t < NPG * FF; t += 256) acc[t] = 0.0f;
    if (tid < NPG) degl[tid] = 0.0f;
    __syncthreads();

    for (int i = tid; i < 2 * MM; i += 256) {
        int e = ebase + i;
        float w = ew ? ew[e] : 1.0f;
        if (w != 0.0f) atomicAdd(&degl[dstv[e] - nbase], w);
    }
    __syncthreads();

    if (tid < NPG) dinvl[tid] = rsqrtf(degl[tid] + 1.0f);  // +1 self loop
    __syncthreads();

    const int wv = tid >> 5, lane = tid & 31;
    for (int i = wv; i < 2 * MM; i += 8) {
        int e = ebase + i;
        int s = srcv[e], d = dstv[e];
        float w = ew ? ew[e] : 1.0f;
        float norm = dinvl[s - nbase] * w * dinvl[d - nbase];
        if (norm == 0.0f) continue;      // pruned edge (uniform per wave)
        const float* hp = h + (size_t)s * FF;
        float* ap = acc + (d - nbase) * FF;
#pragma unroll
        for (int j = 0; j < 4; ++j) {
            int f = lane + 32 * j;
            atomicAdd(&ap[f], norm * hp[f]);
        }
    }
    __syncthreads();

    for (int t = tid; t < NPG * FF; t += 256) {
        int nl = t >> 7, f = t & (FF - 1);
        float di = dinvl[nl];
        float v = acc[t] + di * di * h[(size_t)nbase * FF + t] + bias[f];
        out[(size_t)nbase * FF + t] = relu ? fmaxf(v, 0.0f) : v;
    }
}

// scores[e] = dot(x1[src], x1[dst]) — wave per edge, shfl reduction
__global__ void edge_score_kernel(const float* __restrict__ x1,
                                  const int* __restrict__ srcv,
                                  const int* __restrict__ dstv,
                                  float* __restrict__ scores) {
    int e = blockIdx.x * (blockDim.x >> 5) + (threadIdx.x >> 5);
    if (e >= EE) return;
    int lane = threadIdx.x & 31;
    int s = srcv[e], d = dstv[e];
    const float4* ps = reinterpret_cast<const float4*>(x1 + (size_t)s * FF);
    const float4* pd = reinterpret_cast<const float4*>(x1 + (size_t)d * FF);
    float4 a = ps[lane], b = pd[lane];
    float v = a.x * b.x + a.y * b.y + a.z * b.z + a.w * b.w;
#pragma unroll
    for (int off = 16; off > 0; off >>= 1) v += __shfl_xor(v, off, 32);
    if (lane == 0) scores[e] = v;
}

// ---------------------------------------------------------------------------
// Per-graph top-k over masked edges (src<dst). One WG per graph, 512 threads.
// Rank by score desc, stable by index (matches jnp.lexsort). sel mirrored to
// rev[e]. Writes sampled (straight to d_out) and ew = scores * sampled.
// ---------------------------------------------------------------------------
__global__ void topk_kernel(const float* __restrict__ scores,
                            const int* __restrict__ srcv,
                            const int* __restrict__ dstv,
                            const int* __restrict__ rev,
                            float* __restrict__ ew,
                            float* __restrict__ sampled) {
    __shared__ float s_sc[MM];
    __shared__ int s_cnt;
    int g = blockIdx.x, t = threadIdx.x;
    int e = g * (2 * MM) + t;
    int s = srcv[e], d = dstv[e];
    bool msk = s < d;
    float sc = scores[e];
    float smask = msk ? sc : -INFINITY;
    if (t == 0) s_cnt = 0;
    __syncthreads();
    s_sc[t] = smask;
    if (msk) atomicAdd(&s_cnt, 1);
    __syncthreads();
    int m = s_cnt;
    int k = (m + 1) >> 1;                 // ceil(0.5 * m)
    int rank = 0;
    for (int j = 0; j < MM; ++j) {
        float sj = s_sc[j];
        rank += (sj > smask) || (sj == smask && j < t);
    }
    bool sel = msk && (rank < k);
    float samp = sel ? 1.0f : 0.0f;
    int re = rev[e];
    sampled[e]  = samp;
    sampled[re] = samp;                   // sel | sel[rev]; rev edge unmasked
    ew[e]  = sc * samp;
    ew[re] = scores[re] * samp;
}

// pooled[g][f] = mean over NPG nodes of x3 — one WG (128 thr) per graph
__global__ void pool_kernel(const float* __restrict__ x3,
                            float* __restrict__ pooled) {
    int g = blockIdx.x, f = threadIdx.x;
    const float* p = x3 + (size_t)g * NPG * FF + f;
    float accv = 0.0f;
#pragma unroll 4
    for (int i = 0; i < NPG; ++i) accv += p[(size_t)i * FF];
    pooled[(size_t)g * FF + f] = accv * (1.0f / NPG);
}

// logits[g][c] = dot(hmid[g], lw2[:,c]) + lb2[c]
__global__ void logits_kernel(const float* __restrict__ hmid,
                              const float* __restrict__ lw2,
                              const float* __restrict__ lb2,
                              float* __restrict__ logits) {
    int t = blockIdx.x * blockDim.x + threadIdx.x;
    if (t >= GG * CLS) return;
    int g = t / CLS, c = t % CLS;
    const float* hp = hmid + (size_t)g * FF;
    float accv = lb2[c];
    for (int kk = 0; kk < FF; ++kk) accv += hp[kk] * lw2[(size_t)kk * CLS + c];
    logits[t] = accv;
}

__global__ void logsoftmax_kernel(const float* __restrict__ logits,
                                  float* __restrict__ out) {
    int g = blockIdx.x * blockDim.x + threadIdx.x;
    if (g >= GG) return;
    const float* lp = logits + (size_t)g * CLS;
    float mx = lp[0];
#pragma unroll
    for (int c = 1; c < CLS; ++c) mx = fmaxf(mx, lp[c]);
    float se = 0.0f;
#pragma unroll
    for (int c = 0; c < CLS; ++c) se += __expf(lp[c] - mx);
    float lse = mx + __logf(se);
#pragma unroll
    for (int c = 0; c < CLS; ++c) out[(size_t)g * CLS + c] = lp[c] - lse;
}

__global__ void counts_kernel(float* __restrict__ counts) {
    int g = blockIdx.x * blockDim.x + threadIdx.x;
    if (g < GG) counts[g] = (float)(2 * MM);   // every graph has 2M edges
}

// ---------------------------------------------------------------------------
extern "C" void kernel_launch(void* const* d_in, const int* in_sizes, int n_in,
                              void* d_out, int out_size, void* d_ws, size_t ws_size,
                              hipStream_t stream) {
    const float* x    = (const float*)d_in[0];
    const int*   src  = (const int*)d_in[1];
    const int*   dst  = (const int*)d_in[2];
    const int*   rev  = (const int*)d_in[3];
    /* batch = d_in[4] (structure: node n -> graph n/NPG) */
    const float* W0 = (const float*)d_in[5];  const float* b0 = (const float*)d_in[6];
    const float* W1 = (const float*)d_in[7];  const float* b1 = (const float*)d_in[8];
    const float* W2 = (const float*)d_in[9];  const float* b2 = (const float*)d_in[10];
    const float* lw1 = (const float*)d_in[11]; const float* lb1 = (const float*)d_in[12];
    const float* lw2 = (const float*)d_in[13]; const float* lb2 = (const float*)d_in[14];

    // d_out layout: log_softmax [G*CLS] | sampled [E] | counts [G]
    float* out_logp    = (float*)d_out;
    float* out_sampled = out_logp + (size_t)GG * CLS;
    float* out_counts  = out_sampled + (size_t)EE;

    // Workspace layout (floats) — no zero-init needed anywhere
    float* ws   = (float*)d_ws;
    const size_t NF = (size_t)NN * FF;       // 8,388,608
    float* H    = ws;                        // transformed features (x@W)
    float* X1   = H    + NF;                 // x1, then x2, then x3 (reused)
    float* SC   = X1   + NF;                 // [E] edge scores
    float* EW   = SC   + EE;                 // [E] pruned edge weights
    float* POOL = EW   + EE;                 // [G*F]
    float* HMID = POOL + (size_t)GG * FF;    // [G*F]
    float* LOG  = HMID + (size_t)GG * FF;    // [G*CLS]

    const int gemmBlocks = NN / (16 * 8);      // 512: 8 strips per block
    const int edgeBlocks = (EE * 32) / 256;    // wave-per-edge kernels

    // ---- Layer 1: x1 = relu(gcn(x, W0, b0, ew=1)) ----
    gemm128_wmma_kernel<<<gemmBlocks, 256, 0, stream>>>(x, W0, nullptr, H, NN, 0);
    gcn_layer_kernel<<<GG, 256, 0, stream>>>(H, src, dst, nullptr, b0, X1, 1);

    // ---- Edge scoring + per-graph top-k pruning ----
    edge_score_kernel<<<edgeBlocks, 256, 0, stream>>>(X1, src, dst, SC);
    topk_kernel<<<GG, MM, 0, stream>>>(SC, src, dst, rev, EW, out_sampled);

    // ---- Layer 2: x2 = relu(gcn(x1, W1, b1, ew)) ----
    gemm128_wmma_kernel<<<gemmBlocks, 256, 0, stream>>>(X1, W1, nullptr, H, NN, 0);
    gcn_layer_kernel<<<GG, 256, 0, stream>>>(H, src, dst, EW, b1, X1, 1);  // X1 := x2

    // ---- Layer 3: x3 = relu(gcn(x2, W2, b2, ew)) ----
    gemm128_wmma_kernel<<<gemmBlocks, 256, 0, stream>>>(X1, W2, nullptr, H, NN, 0);
    gcn_layer_kernel<<<GG, 256, 0, stream>>>(H, src, dst, EW, b2, X1, 1);  // X1 := x3

    // ---- Pool + head ----
    pool_kernel<<<GG, FF, 0, stream>>>(X1, POOL);
    gemm128_wmma_kernel<<<GG / (16 * 8), 256, 0, stream>>>(POOL, lw1, lb1, HMID, GG, 1);
    logits_kernel<<<(GG * CLS + 255) / 256, 256, 0, stream>>>(HMID, lw2, lb2, LOG);
    logsoftmax_kernel<<<GG / 256, 256, 0, stream>>>(LOG, out_logp);
    counts_kernel<<<GG / 256, 256, 0, stream>>>(out_counts);
}